// RGCNConv_65687229825994
// MI455X (gfx1250) — compile-verified
//
#include <hip/hip_runtime.h>

typedef __attribute__((ext_vector_type(2))) float v2f;
typedef __attribute__((ext_vector_type(8))) float v8f;

#define DIN  128
#define DOUT 128

// ---------------------------------------------------------------------------
// Zero workspace (sums [N*128] + counts [N], contiguous)
// ---------------------------------------------------------------------------
__global__ void zero_f32(float* __restrict__ p, int n) {
    int i = blockIdx.x * blockDim.x + threadIdx.x;
    if (i < n) p[i] = 0.0f;
}

// ---------------------------------------------------------------------------
// Edge scatter: one wave (32 lanes) per edge. Each lane gathers a float4
// (16 B) of the source node's row (coalesced within the row) and atomically
// adds it into sums[dst]. Lane 0 bumps the in-degree count.
// ---------------------------------------------------------------------------
__global__ void scatter_sum(const float* __restrict__ x,
                            const int*  __restrict__ src,
                            const int*  __restrict__ dst,
                            float* __restrict__ sums,
                            float* __restrict__ cnts,
                            int E) {
    int t    = blockIdx.x * blockDim.x + threadIdx.x;
    int e    = t >> 5;
    int lane = t & 31;
    if (e >= E) return;
    int s = src[e];
    int d = dst[e];
    const float4 v = ((const float4*)(x + (size_t)s * DIN))[lane];
    float* o = sums + (size_t)d * DIN + lane * 4;
    atomicAdd(o + 0, v.x);
    atomicAdd(o + 1, v.y);
    atomicAdd(o + 2, v.z);
    atomicAdd(o + 3, v.w);
    if (lane == 0) atomicAdd(cnts + d, 1.0f);
}

// ---------------------------------------------------------------------------
// WMMA GEMM: out[n][o] (+)= scale(n) * A[n][:] . W[o][:]  (+ bias[o])
//   REL=false : root transform, bias added, overwrite out
//   REL=true  : relation transform, A scaled by 1/count (mean), out += result
//
// One wave handles a 16-row strip x all 128 output columns using
// V_WMMA_F32_16X16X4_F32, accumulating in 8x v8f.
//
// W^T is staged in LDS in K-PAIR layout: wl2[p*DOUT + o] = {W[o][2p], W[o][2p+1]}
// so each B fragment is one aligned 8-byte ds load landing directly in the
// even VGPR pair WMMA wants. All 8 B fragments of a k-step are fetched into
// distinct registers BEFORE the WMMA chain so the scheduler can clause the
// LDS loads and retire WMMAs against graduated dscnt waits.
// ---------------------------------------------------------------------------
template <bool REL>
__global__ void gemm_wmma(const float* __restrict__ A,
                          const float* __restrict__ W,
                          const float* __restrict__ bias,
                          const float* __restrict__ cnts,
                          float* __restrict__ out,
                          int nrows) {
    __shared__ v2f wl2[(DIN / 2) * DOUT];   // 64 KB: [k-pair][o]

    // Cooperative, coalesced load of W with transpose into LDS pair layout.
    for (int idx = threadIdx.x; idx < DOUT * (DIN / 4); idx += blockDim.x) {
        int o  = idx >> 5;             // output row of W
        int i4 = (idx & 31) << 2;      // starting input col (multiple of 4)
        float4 v = ((const float4*)(W + (size_t)o * DIN))[idx & 31];
        v2f lo; lo[0] = v.x; lo[1] = v.y;
        v2f hi; hi[0] = v.z; hi[1] = v.w;
        wl2[((i4 >> 1) + 0) * DOUT + o] = lo;   // K = i4, i4+1
        wl2[((i4 >> 1) + 1) * DOUT + o] = hi;   // K = i4+2, i4+3
    }
    __syncthreads();

    const int wave  = threadIdx.x >> 5;          // 0..7
    const int lane  = threadIdx.x & 31;
    const int strip = blockIdx.x * 8 + wave;     // 16-row strip index
    const int row0  = strip * 16;
    if (row0 >= nrows) return;

    const int m    = lane & 15;                  // A-frag row within strip
    const int half = lane >> 4;
    const int kh   = half * 2;                   // A/B frag K offset (0 or 2)
    const int n    = lane & 15;                  // B / C-D column

    // Mean scaling for relation GEMM (applied to A fragments).
    float scale = 1.0f;
    if (REL) {
        float c = cnts[row0 + m];
        scale = (c > 0.0f) ? (1.0f / c) : 0.0f;
    }

    // Accumulator init: zeros (REL) or broadcast bias along columns (root).
    v8f acc[8];
#pragma unroll
    for (int ot = 0; ot < 8; ++ot) {
        float c0 = REL ? 0.0f : bias[ot * 16 + n];
#pragma unroll
        for (int r = 0; r < 8; ++r) acc[ot][r] = c0;
    }

    const float* arow = A + (size_t)(row0 + m) * DIN + kh;

    for (int k0 = 0; k0 < DIN; k0 += 4) {
        v2f a = *(const v2f*)(arow + k0);        // A[m][k0+kh], A[m][k0+kh+1]
        if (REL) { a[0] *= scale; a[1] *= scale; }
        const int p = (k0 >> 1) + half;          // k-pair index = (k0+kh)/2

        // Fetch all 8 B fragments first (4x ds_load_2addr_b64, clause-able).
        v2f b[8];
#pragma unroll
        for (int ot = 0; ot < 8; ++ot)
            b[ot] = wl2[p * DOUT + ot * 16 + n]; // {B[kb][col], B[kb+1][col]}

#pragma unroll
        for (int ot = 0; ot < 8; ++ot)
            acc[ot] = __builtin_amdgcn_wmma_f32_16x16x4_f32(
                false, a, false, b[ot], (short)0, acc[ot], false, false);
    }

    // Store: C/D layout -> row = r + 8*half, col = ot*16 + n.
#pragma unroll
    for (int ot = 0; ot < 8; ++ot) {
#pragma unroll
        for (int r = 0; r < 8; ++r) {
            float* ptr = out + (size_t)(row0 + r + 8 * half) * DOUT + ot * 16 + n;
            if (REL) *ptr += acc[ot][r];
            else     *ptr  = acc[ot][r];
        }
    }
}

// ---------------------------------------------------------------------------
// Launch: 3 root GEMMs, then per relation {zero, scatter, mean-GEMM-accum}.
// ---------------------------------------------------------------------------
extern "C" void kernel_launch(void* const* d_in, const int* in_sizes, int n_in,
                              void* d_out, int out_size, void* d_ws, size_t ws_size,
                              hipStream_t stream) {
    (void)n_in; (void)out_size; (void)ws_size;

    const int N = in_sizes[0] / DIN;     // 100000
    const int E = in_sizes[10];          // 500000

    const float* x[3]     = {(const float*)d_in[0], (const float*)d_in[3], (const float*)d_in[6]};
    const float* wroot[3] = {(const float*)d_in[1], (const float*)d_in[4], (const float*)d_in[7]};
    const float* broot[3] = {(const float*)d_in[2], (const float*)d_in[5], (const float*)d_in[8]};
    const float* wrel[4]  = {(const float*)d_in[9], (const float*)d_in[12],
                             (const float*)d_in[15], (const float*)d_in[18]};
    const int* srcj[4] = {(const int*)d_in[10], (const int*)d_in[13],
                          (const int*)d_in[16], (const int*)d_in[19]};
    const int* dstj[4] = {(const int*)d_in[11], (const int*)d_in[14],
                          (const int*)d_in[17], (const int*)d_in[20]};

    float* out  = (float*)d_out;
    float* sums = (float*)d_ws;                     // N*128 floats
    float* cnts = sums + (size_t)N * DIN;           // N floats (contiguous)

    const dim3 blk(256);
    const int  strips = (N + 15) / 16;
    const dim3 ggrid((strips + 7) / 8);

    // Root transforms (overwrite d_out, clearing poison).
    for (int t = 0; t < 3; ++t) {
        gemm_wmma<false><<<ggrid, blk, 0, stream>>>(
            x[t], wroot[t], broot[t], nullptr, out + (size_t)t * N * DOUT, N);
    }

    // EDGE_TYPES = [(0,1),(1,2),(2,0),(0,2)]
    const int st[4] = {0, 1, 2, 0};
    const int dt[4] = {1, 2, 0, 2};

    const int  zn = N * DIN + N;
    const dim3 zgrid((zn + 255) / 256);
    const dim3 sgrid((unsigned)(((size_t)E * 32 + 255) / 256));

    for (int j = 0; j < 4; ++j) {
        zero_f32<<<zgrid, blk, 0, stream>>>(sums, zn);
        scatter_sum<<<sgrid, blk, 0, stream>>>(x[st[j]], srcj[j], dstj[j], sums, cnts, E);
        gemm_wmma<true><<<ggrid, blk, 0, stream>>>(
            sums, wrel[j], nullptr, cnts, out + (size_t)dt[j] * N * DOUT, N);
    }
}